// CTCLayer_84971632984347
// MI455X (gfx1250) — compile-verified
//
#include <hip/hip_runtime.h>
#include <cstdint>
#include <cstddef>

// ---------------------------------------------------------------------------
// CTC batch cost, forward-only, prob-domain recurrence with windowed rescale.
// B=256 blocks x 32 threads (1 wave32 per batch item).
// y_pred rows streamed into LDS via CDNA5 Tensor Data Mover (tensor_load_to_lds),
// triple-buffered 8-row (32KB) chunks, synced with s_wait_tensorcnt.
// Renormalization (wave butterfly sum) only every RESCALE_ROWS rows to keep
// the serial cross-lane reduction off the per-row critical path.
// ---------------------------------------------------------------------------

namespace {
constexpr int Bn = 256;
constexpr int Tn = 256;
constexpr int Cn = 1024;
constexpr int Ln = 32;
constexpr int Sn = 2 * Ln + 1;           // 65 extended states
constexpr int CHUNK_ROWS = 8;            // rows (time steps) per TDM tile
constexpr int NBUF = 3;                  // triple buffering
constexpr int NCHUNK = Tn / CHUNK_ROWS;  // 32
constexpr int RESCALE_MASK = 3;          // rescale when (r & 3) == 3
constexpr float EPSF = 1e-7f;
constexpr unsigned CHUNK_BYTES = CHUNK_ROWS * Cn * 4;
constexpr size_t SMEM_ROWS_FLOATS = (size_t)NBUF * CHUNK_ROWS * Cn;
constexpr size_t SMEM_BYTES = SMEM_ROWS_FLOATS * 4 + Ln * 4 + (Sn + 3) * 4;
}  // namespace

typedef unsigned int v4u __attribute__((ext_vector_type(4)));
typedef int v4i __attribute__((ext_vector_type(4)));
typedef int v8i __attribute__((ext_vector_type(8)));

// Issue one TDM load: CHUNK_ROWS x Cn f32 tile (contiguous rows, stride Cn)
// from global 'gaddr_bytes' into LDS byte offset 'lds_addr'.
__device__ __forceinline__ void tdm_load_rows(uint32_t lds_addr,
                                              uint64_t gaddr_bytes) {
  // D# group 0: count=1 | lds_addr | global_addr[56:0] | type=2
  v4u g0;
  g0.x = 1u;                                         // count=1 (valid user D#)
  g0.y = lds_addr;                                   // LDS byte address
  g0.z = (uint32_t)(gaddr_bytes & 0xFFFFFFFFull);    // global_addr lo
  g0.w = (uint32_t)((gaddr_bytes >> 32) & 0x1FFFFFFull) | (2u << 30);  // hi|type

  // D# group 1: data_size=4B, tensor_dim0=Cn, tensor_dim1=CHUNK_ROWS,
  // tile_dim0=Cn, tile_dim1=CHUNK_ROWS, tensor_dim0_stride=Cn (elements).
  v8i g1;
  g1[0] = (int)(2u << 16);                                      // data_size=2 (4B)
  g1[1] = (int)(((uint32_t)Cn & 0xFFFFu) << 16);                // tdim0 lo16
  g1[2] = (int)((((uint32_t)Cn >> 16) & 0xFFFFu) |
                ((uint32_t)CHUNK_ROWS << 16));                  // tdim0 hi | tdim1 lo
  g1[3] = (int)(((uint32_t)Cn & 0xFFFFu) << 16);                // tdim1 hi(0)|tile_dim0
  g1[4] = (int)CHUNK_ROWS;                                      // tile_dim1 | tile_dim2=0
  g1[5] = (int)Cn;                                              // stride0 lo32
  g1[6] = 0;                                                    // stride0 hi | stride1 lo
  g1[7] = 0;                                                    // stride1 hi

  // Groups 2/3: degenerate higher dims (len 1, tiles unused).
  v4i g2; g2[0] = 1; g2[1] = 1; g2[2] = 0; g2[3] = 0;
  v4i g3; g3[0] = 0; g3[1] = (int)(1u << 16); g3[2] = 0; g3[3] = 0;

#if defined(__clang_major__) && (__clang_major__ >= 23)
  v8i g4 = {0, 0, 0, 0, 0, 0, 0, 0};
  __builtin_amdgcn_tensor_load_to_lds(g0, g1, g2, g3, g4, 0);
#else
  __builtin_amdgcn_tensor_load_to_lds(g0, g1, g2, g3, 0);
#endif
}

__global__ void __launch_bounds__(32) ctc_fwd_kernel(
    const int* __restrict__ y_true, const float* __restrict__ y_pred,
    float* __restrict__ out) {
  extern __shared__ float smem[];
  float* rows = smem;                                    // NBUF*CHUNK_ROWS*Cn f32
  int* labs = (int*)(rows + SMEM_ROWS_FLOATS);           // Ln ints
  float* afin = (float*)(labs + Ln);                     // Sn (+pad) f32

  const int b = blockIdx.x;
  const int lane = threadIdx.x;

  if (lane < Ln) labs[lane] = y_true[b * Ln + lane];
  __syncthreads();

  const int myl = (lane < Ln) ? labs[lane] : 0;
  const unsigned long long bal = __ballot((lane < Ln) && (myl != 0));
  const int len = (int)__popcll(bal);  // label length (count of nonzero)

  // Per-lane state setup: lane l owns states s = l, l+32, l+64.
  int clsIdx[3];
  float skipm[3], vmask[3], initm[3];
#pragma unroll
  for (int k = 0; k < 3; ++k) {
    const int s = lane + 32 * k;
    const bool valid = (s < Sn);
    int cls = Cn - 1;  // blank
    float sk = 0.0f;
    if (valid && (s & 1)) {
      const int i = (s - 1) >> 1;
      const int li = labs[i];
      const int pr = (i >= 1) ? labs[i - 1] : -1;
      cls = li;
      sk = ((li != (Cn - 1)) && (li != pr)) ? 1.0f : 0.0f;  // allow_skip
    }
    if (!valid) cls = 0;
    clsIdx[k] = cls;
    skipm[k] = sk;
    vmask[k] = valid ? 1.0f : 0.0f;
    initm[k] = (valid && (s < 2)) ? 1.0f : 0.0f;
  }

  const uint32_t lds_rows_base = (uint32_t)(uintptr_t)rows;
  const uint64_t gbase = (uint64_t)(uintptr_t)(y_pred + (size_t)b * Tn * Cn);

  // Prologue: fill the pipeline with NBUF chunks.
#pragma unroll
  for (int q = 0; q < NBUF; ++q)
    tdm_load_rows(lds_rows_base + (uint32_t)q * CHUNK_BYTES,
                  gbase + (uint64_t)q * CHUNK_BYTES);

  float alpha[3] = {0.0f, 0.0f, 0.0f};
  float acc = 0.0f;  // accumulated log-scale

  for (int c = 0; c < NCHUNK; ++c) {
    // Wait until chunk c has landed (in-order TENSORcnt: <= NBUF-1 outstanding).
    __builtin_amdgcn_s_wait_tensorcnt((short)(NBUF - 1));
    asm volatile("" ::: "memory");
    const float* buf = rows + (size_t)(c % NBUF) * CHUNK_ROWS * Cn;

    float pv[3];
#pragma unroll
    for (int k = 0; k < 3; ++k) pv[k] = buf[clsIdx[k]] + EPSF;

#pragma unroll
    for (int r = 0; r < CHUNK_ROWS; ++r) {
      // Preload next row's gathered probs to hide LDS latency.
      float pn[3] = {pv[0], pv[1], pv[2]};
      if (r + 1 < CHUNK_ROWS) {
#pragma unroll
        for (int k = 0; k < 3; ++k) pn[k] = buf[(r + 1) * Cn + clsIdx[k]] + EPSF;
      }

      const int t = c * CHUNK_ROWS + r;
      if (t == 0) {
#pragma unroll
        for (int k = 0; k < 3; ++k) alpha[k] = initm[k] * pv[k];
      } else {
        // alpha[s-1], alpha[s-2] via lane rotations (state s = lane + 32k).
        const float sh1_0 = __shfl(alpha[0], (lane + 31) & 31, 32);
        const float sh1_1 = __shfl(alpha[1], (lane + 31) & 31, 32);
        const float sh2_0 = __shfl(alpha[0], (lane + 30) & 31, 32);
        const float sh2_1 = __shfl(alpha[1], (lane + 30) & 31, 32);
        float am1[3], am2[3];
        am1[0] = (lane >= 1) ? sh1_0 : 0.0f;
        am1[1] = (lane >= 1) ? sh1_1 : sh1_0;
        am1[2] = (lane >= 1) ? 0.0f : sh1_1;   // only lane0 (s=64) meaningful
        am2[0] = (lane >= 2) ? sh2_0 : 0.0f;
        am2[1] = (lane >= 2) ? sh2_1 : sh2_0;
        am2[2] = (lane >= 2) ? 0.0f : sh2_1;
#pragma unroll
        for (int k = 0; k < 3; ++k) {
          const float ssum = alpha[k] + am1[k] + skipm[k] * am2[k];
          alpha[k] = vmask[k] * (pv[k] * ssum);
        }
      }

      // Windowed renormalization: only every 4th row, so the serial 5-stage
      // butterfly + rcp + log sit off the per-row critical path. The linear
      // recurrence makes deferred scaling exact; f32 range easily covers the
      // ~2^-40..2^-66 drift per 4-row window.
      if ((r & RESCALE_MASK) == RESCALE_MASK) {
        float part = alpha[0] + alpha[1] + alpha[2];
#pragma unroll
        for (int off = 16; off >= 1; off >>= 1)
          part += __shfl_xor(part, off, 32);
        const float inv = 1.0f / part;
        alpha[0] *= inv;
        alpha[1] *= inv;
        alpha[2] *= inv;
        acc += logf(part);
      }

      pv[0] = pn[0]; pv[1] = pn[1]; pv[2] = pn[2];
    }

    // Buffer (c % NBUF) fully consumed: refill with chunk c+NBUF.
    asm volatile("" ::: "memory");
    const int nc = c + NBUF;
    if (nc < NCHUNK)
      tdm_load_rows(lds_rows_base + (uint32_t)(nc % NBUF) * CHUNK_BYTES,
                    gbase + (uint64_t)nc * CHUNK_BYTES);
  }

  // loss = -( log(alpha[2*len] + alpha[2*len-1]) + acc )
  // (alpha is sum-normalized at this point: T=256 ends on a rescale row.)
#pragma unroll
  for (int k = 0; k < 3; ++k) {
    const int s = lane + 32 * k;
    if (s < Sn) afin[s] = alpha[k];
  }
  __syncthreads();
  if (lane == 0) {
    const int ib = 2 * len;
    const int il = 2 * len - 1;
    const float fa = afin[ib];
    const float fb = (il >= 0) ? afin[il] : 0.0f;
    out[b] = -(logf(fa + fb) + acc);
  }
}

extern "C" void kernel_launch(void* const* d_in, const int* in_sizes, int n_in,
                              void* d_out, int out_size, void* d_ws,
                              size_t ws_size, hipStream_t stream) {
  (void)in_sizes; (void)n_in; (void)out_size; (void)d_ws; (void)ws_size;
  const int* y_true = (const int*)d_in[0];
  const float* y_pred = (const float*)d_in[1];
  float* out = (float*)d_out;
  ctc_fwd_kernel<<<Bn, 32, (unsigned)SMEM_BYTES, stream>>>(y_true, y_pred, out);
}